// Linear_35794257445488
// MI455X (gfx1250) — compile-verified
//
#include <hip/hip_runtime.h>
#include <math.h>

#define NAG 4096
#define TILES (NAG / 16)

typedef __attribute__((ext_vector_type(2))) float v2f;
typedef __attribute__((ext_vector_type(8))) float v8f;

// Constants from the reference
#define PED_SPEED   1.5f
#define ROBOT_SPEED 1.5f
#define K_ATTR      2.0f
#define ALPHA       10.0f
#define PED_RADIUS  0.3f
#define PED_MASS    60.0f
#define BETTA       0.71f
#define DT          0.4f
#define A_COST      4.0f
#define B_COST      1.2f
#define E_COST      0.001f
#define EPS         1e-8f
#define LOG2E       1.44269504089f

// --------------------------------------------------------------------------
// Kernel 1: all-pairs repulsion (WMMA distance tiles) + attraction + pose
// propagation. One wave per 16-row tile; 8 waves/block, 32 blocks = 256 tiles.
// --------------------------------------------------------------------------
__global__ __launch_bounds__(256) void sf_forces_kernel(
    const float* __restrict__ state,   // N x 4 : x,y,vx,vy
    const float* __restrict__ goals,   // N x 2
    float* __restrict__ out)           // N x 4 propagated state
{
  __shared__ float4 sPos[NAG];         // (x, y, x^2+y^2, 1)  = 64 KB

  const int tid = threadIdx.x;

  // Stage all agent positions into LDS (each thread stages 16 agents)
  for (int a = tid; a < NAG; a += 256) {
    float x = state[a * 4 + 0];
    float y = state[a * 4 + 1];
    sPos[a] = make_float4(x, y, x * x + y * y, 1.0f);
  }
  __syncthreads();

  const int lane  = tid & 31;
  const int wave  = tid >> 5;
  // tile is uniform across the wave; tell the compiler so the diagonal guard
  // and tile addressing become scalar (s_cmp/s_cbranch, SALU addressing).
  const int tile  = __builtin_amdgcn_readfirstlane(blockIdx.x * 8 + wave);
  const int ibase = tile * 16;
  const int half  = lane >> 4;   // 0: lanes 0-15, 1: lanes 16-31
  const int col   = lane & 15;

  // A operand (16x4 f32): row i = [x_i, y_i, s_i, 1]
  // lanes 0-15 hold K=0,1 (VGPR0,1); lanes 16-31 hold K=2,3.
  float4 pa = sPos[ibase + col];
  v2f A;
  A[0] = half ? pa.z : pa.x;
  A[1] = half ? 1.0f : pa.y;

  // Accumulators: Sw = sum_j w_ij, Sx = sum_j w_ij*x_j, Sy = sum_j w_ij*y_j
  float Sw[8], Sx[8], Sy[8];
#pragma unroll
  for (int v = 0; v < 8; ++v) { Sw[v] = 0.0f; Sx[v] = 0.0f; Sy[v] = 0.0f; }

  // ALPHA*exp((2R - d)/BETTA) = exp2(C0 - C1*d)
  const float C1 = LOG2E / BETTA;
  const float C0 = 3.32192809489f /*log2(10)*/ + (2.0f * PED_RADIUS) * C1;

  // C seed: fold the reference's "+EPS" (inside the sqrt) into the WMMA
  // accumulator: D = A*B + EPS = |p_i - p_j|^2 + EPS.
  v8f Cseed;
#pragma unroll
  for (int v = 0; v < 8; ++v) Cseed[v] = EPS;

#pragma unroll 2
  for (int jt = 0; jt < TILES; ++jt) {
    const int jbase = jt * 16;
    const float4 pb = sPos[jbase + col];

    // B operand (4x16 f32): col j = [-2x_j, -2y_j, 1, s_j]
    // VGPR0: K=0 (lanes 0-15) / K=2 (lanes 16-31); VGPR1: K=1 / K=3.
    v2f B;
    B[0] = half ? 1.0f : (-2.0f * pb.x);
    B[1] = half ? pb.z : (-2.0f * pb.y);

    // D[M][N] = s_i + s_j - 2 p_i.p_j + EPS = |p_i - p_j|^2 + EPS
    v8f C = __builtin_amdgcn_wmma_f32_16x16x4_f32(false, A, false, B,
                                                  (short)0, Cseed,
                                                  false, false);

    float w[8];
#pragma unroll
    for (int v = 0; v < 8; ++v) {
      float d2 = C[v];
      float r  = __builtin_amdgcn_rsqf(d2);           // 1/d
      float d  = d2 * r;                              // d
      float e  = __builtin_amdgcn_exp2f(C0 - C1 * d); // ALPHA*exp((2R-d)/B)
      w[v] = e * r;                                   // mag / d
    }

    // Diagonal occurs only in the tile where jt == tile; both sides are
    // wave-uniform scalars -> pure s_cmp/s_cbranch, skipped 255/256 times.
    if (jt == tile) {
      const int jcol = jbase + col;
#pragma unroll
      for (int v = 0; v < 8; ++v)
        if (jcol == ibase + half * 8 + v) w[v] = 0.0f;
    }

#pragma unroll
    for (int v = 0; v < 8; ++v) {
      Sw[v] += w[v];
      Sx[v] += w[v] * pb.x;
      Sy[v] += w[v] * pb.y;
    }
  }

  // Reduce across the 16 lanes of each half (xor masks 1,2,4,8 stay in-half)
#pragma unroll
  for (int m = 1; m < 16; m <<= 1) {
#pragma unroll
    for (int v = 0; v < 8; ++v) {
      Sw[v] += __shfl_xor(Sw[v], m, 32);
      Sx[v] += __shfl_xor(Sx[v], m, 32);
      Sy[v] += __shfl_xor(Sy[v], m, 32);
    }
  }

  // 16 lanes per wave finish one row each: lanes 0-7 -> rows ibase+0..7,
  // lanes 16-23 -> rows ibase+8..15.
  if ((lane & 8) == 0) {
    const int v = lane & 7;
    float sw = Sw[0], sx = Sx[0], sy = Sy[0];
#pragma unroll
    for (int k = 1; k < 8; ++k)
      if (v == k) { sw = Sw[k]; sx = Sx[k]; sy = Sy[k]; }

    const int i = ibase + half * 8 + v;
    const float4 pi4 = sPos[i];
    const float vx = state[i * 4 + 2];
    const float vy = state[i * 4 + 3];

    // Repulsion: sum_j w_ij * (p_i - p_j)
    float rfx = pi4.x * sw - sx;
    float rfy = pi4.y * sw - sy;

    // Attraction
    float gx = goals[i * 2 + 0] - pi4.x;
    float gy = goals[i * 2 + 1] - pi4.y;
    float dg = sqrtf(gx * gx + gy * gy);
    float inv = 1.0f / (dg + EPS);
    float dsp = (i == 0) ? ROBOT_SPEED : PED_SPEED;
    float afx = K_ATTR * (dsp * gx * inv - vx) * PED_MASS;
    float afy = K_ATTR * (dsp * gy * inv - vy) * PED_MASS;

    // Pose propagation
    float Fx = rfx + afx, Fy = rfy + afy;
    float vnx = vx + Fx * (DT / PED_MASS);
    float vny = vy + Fy * (DT / PED_MASS);
    float sp = sqrtf(vnx * vnx + vny * vny);
    float sc = fminf(1.0f, PED_SPEED / (sp + EPS));
    vnx *= sc; vny *= sc;

    out[i * 4 + 0] = pi4.x + vnx * DT;
    out[i * 4 + 1] = pi4.y + vny * DT;
    out[i * 4 + 2] = vnx;
    out[i * 4 + 3] = vny;
  }
}

// --------------------------------------------------------------------------
// Kernel 2: cost (depends on propagated robot pose = out row 0) + concat copy
// --------------------------------------------------------------------------
__global__ __launch_bounds__(256) void sf_cost_kernel(
    const float* __restrict__ state,
    const float* __restrict__ cost_in,
    const float* __restrict__ stacked_in,
    const float* __restrict__ goals,
    const float* __restrict__ rip,     // robot_init_pose (2)
    float* __restrict__ out_all)       // [state N*4 | cost N | stacked 2N*4]
{
  const int i = blockIdx.x * blockDim.x + threadIdx.x;
  if (i >= NAG) return;

  const float* st       = out_all;             // propagated state, N x 4
  float* cost_out       = out_all + NAG * 4;
  float* stacked_out    = out_all + NAG * 5;

  const float rx = st[0], ry = st[1];          // robot pose (row 0)
  const float gdx = goals[0] - rip[0];
  const float gdy = goals[1] - rip[1];
  const float PG = (gdx * (rx - rip[0]) + gdy * (ry - rip[1])) /
                   (sqrtf(gdx * gdx + gdy * gdy) + E_COST);

  const float dx = st[i * 4 + 0] - rx;
  const float dy = st[i * 4 + 1] - ry;
  const float dist = sqrtf(dx * dx + dy * dy + E_COST);
  float blame = __builtin_amdgcn_exp2f(-dist * (LOG2E / B_COST));
  if (i == 0) blame = 0.0f;

  cost_out[i] = cost_in[i] + (-A_COST * PG + blame);

#pragma unroll
  for (int k = 0; k < 4; ++k) {
    stacked_out[i * 4 + k]           = stacked_in[i * 4 + k]; // old stacked
    stacked_out[(NAG + i) * 4 + k]   = state[i * 4 + k];      // append state
  }
}

// --------------------------------------------------------------------------
extern "C" void kernel_launch(void* const* d_in, const int* in_sizes, int n_in,
                              void* d_out, int out_size, void* d_ws, size_t ws_size,
                              hipStream_t stream) {
  const float* state   = (const float*)d_in[0];  // N*4
  const float* cost    = (const float*)d_in[1];  // N
  const float* stacked = (const float*)d_in[2];  // N*4
  const float* goals   = (const float*)d_in[3];  // N*2
  const float* rip     = (const float*)d_in[4];  // 2
  float* out = (float*)d_out;

  sf_forces_kernel<<<TILES / 8, 256, 0, stream>>>(state, goals, out);
  sf_cost_kernel<<<NAG / 256, 256, 0, stream>>>(state, cost, stacked, goals,
                                                rip, out);
}